// ODEDecoder_429496730072
// MI455X (gfx1250) — compile-verified
//
#include <hip/hip_runtime.h>
#include <hip/hip_bf16.h>
#include <math.h>

// ---------------------------------------------------------------------------
// Neural-ODE + MLP decoder for MI455X (gfx1250, wave32, WMMA f32 16x16x4).
//
// Block = 8 waves, 32 batch rows. z state (32x128) lives in registers in the
// WMMA C/D layout (each wave owns N-tile `wave` for both 16-row M halves).
// Every B-matrix (weight) fragment load is shared across the two M halves:
// one B load -> two v_wmma_f32_16x16x4_f32.  LDS staging (zs 16KB, hs 32KB)
// provides the transposed A-operand striping.  Fixed-step RK4, dt = grid dt.
// ---------------------------------------------------------------------------

typedef float v2f __attribute__((ext_vector_type(2)));
typedef float v8f __attribute__((ext_vector_type(8)));

#define LATENT   128
#define HIDDEN   256
#define OUT_DIM  128
#define DEC_H    64
#define T_STEPS  100
#define BATCH    4096
#define TILE_M   32
#define NWAVES   8
#define BLOCK_T  (NWAVES * 32)

// Dual-M-tile GEMM: D{0,1}(16x16) = A{rows 0-15 / 16-31} * B + bias.
// B fragments are loaded ONCE and feed both WMMAs (2x arithmetic intensity).
__device__ __forceinline__ void tile_matmul_pair(
    const float* __restrict__ A_lds, int lda,
    const float* __restrict__ Wg, int ldw,
    int nb, int K, const float* __restrict__ bias,
    int lm, int half, v8f& d0, v8f& d1) {
    const float bv = bias[nb + lm];
#pragma unroll
    for (int r = 0; r < 8; ++r) { d0[r] = bv; d1[r] = bv; }

    const float* a0 = A_lds + lm * lda + 2 * half;          // rows 0..15
    const float* a1 = A_lds + (16 + lm) * lda + 2 * half;   // rows 16..31
    const float* wc = Wg + nb + lm + 2 * half * ldw;
#pragma unroll 4
    for (int kb = 0; kb < K; kb += 4) {
        v2f b, x0, x1;
        b.x  = wc[kb * ldw];
        b.y  = wc[kb * ldw + ldw];
        x0.x = a0[kb];  x0.y = a0[kb + 1];
        x1.x = a1[kb];  x1.y = a1[kb + 1];
        d0 = __builtin_amdgcn_wmma_f32_16x16x4_f32(false, x0, false, b,
                                                   (short)0, d0, false, false);
        d1 = __builtin_amdgcn_wmma_f32_16x16x4_f32(false, x1, false, b,
                                                   (short)0, d1, false, false);
    }
}

// Single-tile GEMM for the small decoder layers (one tile per wave).
__device__ __forceinline__ v8f tile_matmul(
    const float* __restrict__ A_lds, int lda, int mi,
    const float* __restrict__ Wg, int ldw,
    int nb, int K, const float* __restrict__ bias,
    int lm, int half) {
    v8f acc;
    const float bv = bias[nb + lm];
#pragma unroll
    for (int r = 0; r < 8; ++r) acc[r] = bv;

    const float* arow = A_lds + (mi * 16 + lm) * lda + 2 * half;
    const float* wcol = Wg + nb + lm + 2 * half * ldw;
#pragma unroll 4
    for (int kb = 0; kb < K; kb += 4) {
        v2f a, b;
        a.x = arow[kb];
        a.y = arow[kb + 1];
        b.x = wcol[kb * ldw];
        b.y = wcol[kb * ldw + ldw];
        acc = __builtin_amdgcn_wmma_f32_16x16x4_f32(false, a, false, b,
                                                    (short)0, acc, false, false);
    }
    return acc;
}

__device__ __forceinline__ void store_tile_lds(float* __restrict__ S, int lds_,
                                               int mi, int nb, const v8f& v,
                                               int lm, int half) {
#pragma unroll
    for (int r = 0; r < 8; ++r)
        S[(mi * 16 + r + 8 * half) * lds_ + nb + lm] = v[r];
}

__global__ __launch_bounds__(BLOCK_T) void node_rk4_decode_kernel(
    const float* __restrict__ z0, const float* __restrict__ times,
    const float* __restrict__ W1, const float* __restrict__ b1,
    const float* __restrict__ W2, const float* __restrict__ b2,
    const float* __restrict__ dW1, const float* __restrict__ db1,
    const float* __restrict__ dW2, const float* __restrict__ db2,
    const float* __restrict__ dW3, const float* __restrict__ db3,
    float* __restrict__ out) {
    __shared__ float zs[TILE_M * LATENT];   // 16 KB — WMMA A staging (state)
    __shared__ float hs[TILE_M * HIDDEN];   // 32 KB — hidden staging (decoder reuses)

    const int tid  = threadIdx.x;
    const int wave = tid >> 5;
    const int lane = tid & 31;
    const int lm   = lane & 15;
    const int half = lane >> 4;
    const int rowBase = blockIdx.x * TILE_M;
    const int nbS = wave * 16;              // this wave's state N-tile base

    // Per-wave state: N-tile `wave`, both M halves (mi = 0,1).
    v8f z[2], kst[2], acc[2];
#pragma unroll
    for (int mi = 0; mi < 2; ++mi)
#pragma unroll
        for (int r = 0; r < 8; ++r) {
            z[mi][r]   = z0[(rowBase + mi * 16 + r + 8 * half) * LATENT + nbS + lm];
            kst[mi][r] = 0.0f;
        }

    const float cvals[4] = {0.0f, 0.5f, 0.5f, 1.0f};
    const float wvals[4] = {1.0f, 2.0f, 2.0f, 1.0f};

    for (int t = 0; t < T_STEPS; ++t) {
        // ----------------- decode current z -> out[:, t, :] -----------------
#pragma unroll
        for (int mi = 0; mi < 2; ++mi)
            store_tile_lds(zs, LATENT, mi, nbS, z[mi], lm, half);
        __syncthreads();

        const int mi_d = wave >> 2;          // decoder tile ownership
        const int nbd  = (wave & 3) * 16;

        // h1 = relu(z @ dW1 + db1): 32x64, one tile per wave -> hs[:, 0:64)
        {
            v8f h1 = tile_matmul(zs, LATENT, mi_d, dW1, DEC_H, nbd, LATENT, db1, lm, half);
#pragma unroll
            for (int r = 0; r < 8; ++r) h1[r] = fmaxf(h1[r], 0.0f);
            store_tile_lds(hs, HIDDEN, mi_d, nbd, h1, lm, half);
        }
        __syncthreads();

        // h2 = relu(h1 @ dW2 + db2): 32x64 -> hs[:, 64:128)
        {
            v8f h2 = tile_matmul(hs, HIDDEN, mi_d, dW2, DEC_H, nbd, DEC_H, db2, lm, half);
#pragma unroll
            for (int r = 0; r < 8; ++r) h2[r] = fmaxf(h2[r], 0.0f);
            store_tile_lds(hs, HIDDEN, mi_d, DEC_H + nbd, h2, lm, half);
        }
        __syncthreads();

        // y = h2 @ dW3 + db3: 32x128, N-tile `wave`, both M halves, B shared.
        {
            v8f y0, y1;
            tile_matmul_pair(hs + DEC_H, HIDDEN, dW3, OUT_DIM, nbS, DEC_H, db3,
                             lm, half, y0, y1);
#pragma unroll
            for (int r = 0; r < 8; ++r) {
                const int row0 = rowBase + r + 8 * half;
                out[(row0 * T_STEPS + t) * OUT_DIM + nbS + lm]        = y0[r];
                out[((row0 + 16) * T_STEPS + t) * OUT_DIM + nbS + lm] = y1[r];
            }
        }

        if (t == T_STEPS - 1) break;

        // ----------------- advance z with one RK4 step ----------------------
        const float dt = times[t + 1] - times[t];
#pragma unroll
        for (int mi = 0; mi < 2; ++mi)
#pragma unroll
            for (int r = 0; r < 8; ++r) acc[mi][r] = 0.0f;

        for (int s = 0; s < 4; ++s) {
            __syncthreads();  // previous readers of zs/hs are done
            const float c = cvals[s] * dt;
#pragma unroll
            for (int mi = 0; mi < 2; ++mi) {
                v8f zst;
#pragma unroll
                for (int r = 0; r < 8; ++r) zst[r] = fmaf(c, kst[mi][r], z[mi][r]);
                store_tile_lds(zs, LATENT, mi, nbS, zst, lm, half);
            }
            __syncthreads();

            // hidden = tanh(z_stage @ W1 + b1): 32x256, 2 N-pairs per wave,
            // each pair shares B fragments across both M halves.
#pragma unroll
            for (int i = 0; i < 2; ++i) {
                const int nb = (2 * wave + i) * 16;
                v8f h0, h1v;
                tile_matmul_pair(zs, LATENT, W1, HIDDEN, nb, LATENT, b1,
                                 lm, half, h0, h1v);
#pragma unroll
                for (int r = 0; r < 8; ++r) { h0[r] = tanhf(h0[r]); h1v[r] = tanhf(h1v[r]); }
                store_tile_lds(hs, HIDDEN, 0, nb, h0, lm, half);
                store_tile_lds(hs, HIDDEN, 1, nb, h1v, lm, half);
            }
            __syncthreads();

            // k = hidden @ W2 + b2: N-tile `wave`, both M halves (regs).
            tile_matmul_pair(hs, HIDDEN, W2, LATENT, nbS, HIDDEN, b2,
                             lm, half, kst[0], kst[1]);
            const float w = wvals[s];
#pragma unroll
            for (int mi = 0; mi < 2; ++mi)
#pragma unroll
                for (int r = 0; r < 8; ++r) acc[mi][r] = fmaf(w, kst[mi][r], acc[mi][r]);
        }

        const float scale = dt * (1.0f / 6.0f);
#pragma unroll
        for (int mi = 0; mi < 2; ++mi)
#pragma unroll
            for (int r = 0; r < 8; ++r) z[mi][r] = fmaf(scale, acc[mi][r], z[mi][r]);
        __syncthreads();  // hs readers done before next iteration rewrites zs/hs
    }
}

extern "C" void kernel_launch(void* const* d_in, const int* in_sizes, int n_in,
                              void* d_out, int out_size, void* d_ws, size_t ws_size,
                              hipStream_t stream) {
    (void)in_sizes; (void)n_in; (void)out_size; (void)d_ws; (void)ws_size;
    const float* z0    = (const float*)d_in[0];
    const float* times = (const float*)d_in[1];
    const float* W1    = (const float*)d_in[2];
    const float* b1    = (const float*)d_in[3];
    const float* W2    = (const float*)d_in[4];
    const float* b2    = (const float*)d_in[5];
    const float* dW1   = (const float*)d_in[6];
    const float* db1   = (const float*)d_in[7];
    const float* dW2   = (const float*)d_in[8];
    const float* db2   = (const float*)d_in[9];
    const float* dW3   = (const float*)d_in[10];
    const float* db3   = (const float*)d_in[11];
    float* out = (float*)d_out;

    dim3 grid(BATCH / TILE_M);   // 128 blocks
    dim3 block(BLOCK_T);         // 256 threads = 8 waves
    node_rk4_decode_kernel<<<grid, block, 0, stream>>>(
        z0, times, W1, b1, W2, b2, dW1, db1, dW2, db2, dW3, db3, out);
}